// SubglacialDrainageSystem_2491081031887
// MI455X (gfx1250) — compile-verified
//
#include <hip/hip_runtime.h>

#define RR 1024
#define CC 1024
#define NND (RR*CC)            // 1048576 nodes
#define LH (RR*(CC-1))         // horizontal links
#define LV ((RR-1)*CC)         // vertical links
#define LLK (LH+LV)            // 2095104 links

static constexpr float K_FLOW   = 0.0405f;
static constexpr float A_OPEN   = 1.3455e-09f;
static constexpr float C_CLOSE  = 7.11e-24f;
static constexpr float RHOWG    = 9810.0f;    // rho_w * g
static constexpr float INV_DX   = 0.01f;
static constexpr float INV_DX2  = 1.0e-4f;
static constexpr float EPS      = 1e-30f;

#define NPART 1024

__device__ __forceinline__ int iclampi(int v, int lo, int hi) {
    return v < lo ? lo : (v > hi ? hi : v);
}

// ---------------- reduction helpers (wave32) ----------------
__device__ __forceinline__ float block_reduce(float v, float* red) {
    #pragma unroll
    for (int o = 16; o > 0; o >>= 1) v += __shfl_down(v, o, 32);
    const int lane = threadIdx.x & 31;
    const int wid  = threadIdx.x >> 5;
    if (lane == 0) red[wid] = v;
    __syncthreads();
    const int nw = blockDim.x >> 5;
    v = (threadIdx.x < (unsigned)nw) ? red[threadIdx.x] : 0.0f;
    if (wid == 0) {
        #pragma unroll
        for (int o = 16; o > 0; o >>= 1) v += __shfl_down(v, o, 32);
    }
    return v; // valid in thread 0
}

// deterministic block-redundant sum of a 1024-entry partial array, broadcast to all threads
__device__ __forceinline__ float sum_partials(const float* __restrict__ a, float* red, float* bcast) {
    float v = a[threadIdx.x] + a[threadIdx.x + 256] + a[threadIdx.x + 512] + a[threadIdx.x + 768];
    float s = block_reduce(v, red);
    if (threadIdx.x == 0) *bcast = s;
    __syncthreads();
    float r = *bcast;
    __syncthreads();   // protect red/bcast before any subsequent call rewrites them
    return r;
}

// ---------------- 5-point stencil matvec with async-to-LDS halo staging ----------------
// mode 0: y = A*xb, part0[b] = block partial of dot(w, y)           (w = r0)
// mode 1: y = A*xb, part0[b] = partial dot(y, w), part1[b] = partial dot(y, y)   (w = s)
#define BXT 64
#define BYT 16
#define TW  (BXT + 2)          // 66
#define THT (BYT + 2)          // 18
#define TILE_E (TW * THT)      // 1188
__global__ void __launch_bounds__(256)
k_matvec(const float* __restrict__ xb, const float* __restrict__ w,
         float* __restrict__ y, float* __restrict__ part0,
         float* __restrict__ part1, int mode) {
    __shared__ float tile[TILE_E];
    __shared__ float red[32];
    const int tid  = threadIdx.x;
    const int bx   = blockIdx.x & 15;   // 1024/64 = 16 tiles across
    const int by   = blockIdx.x >> 4;   // 64 tiles down
    const int col0 = bx * BXT;
    const int row0 = by * BYT;

    // stage (BYT+2)x(BXT+2) halo tile into LDS via CDNA5 async engine.
    // clamp addresses -> uniform control flow; out-of-grid entries fixed to 0 below.
    for (int e = tid; e < TILE_E; e += 256) {
        int hy = e / TW, hx = e - hy * TW;
        int gr = iclampi(row0 - 1 + hy, 0, RR - 1);
        int gc = iclampi(col0 - 1 + hx, 0, CC - 1);
        unsigned lds_off = (unsigned)(size_t)(&tile[e]);   // low 32 bits = LDS byte offset
        unsigned long long ga = (unsigned long long)(size_t)(xb + (gr << 10) + gc);
        asm volatile("global_load_async_to_lds_b32 %0, %1, off"
                     :: "v"(lds_off), "v"(ga) : "memory");
    }
    asm volatile("s_wait_asynccnt 0" ::: "memory");
    __syncthreads();
    // zero-pad halo entries outside the grid (boundary nodes have fewer neighbors)
    for (int e = tid; e < TILE_E; e += 256) {
        int hy = e / TW, hx = e - hy * TW;
        int gr = row0 - 1 + hy, gc = col0 - 1 + hx;
        if (gr < 0 || gr >= RR || gc < 0 || gc >= CC) tile[e] = 0.0f;
    }
    __syncthreads();

    const int tx  = tid & 63;
    const int ty0 = tid >> 6;           // 0..3, each thread does 4 rows (stride 4)
    const int col = col0 + tx;
    float l0 = 0.0f, l1 = 0.0f;
    #pragma unroll
    for (int rr = 0; rr < 4; ++rr) {
        int ry  = ty0 + rr * 4;         // 0..15
        int row = row0 + ry;
        int i   = (row << 10) + col;
        float c  = tile[(ry + 1) * TW + tx + 1];
        float lf = tile[(ry + 1) * TW + tx];
        float rt = tile[(ry + 1) * TW + tx + 2];
        float up = tile[ ry      * TW + tx + 1];
        float dn = tile[(ry + 2) * TW + tx + 1];
        float deg = (float)((col > 0) + (col < 1023) + (row > 0) + (row < 1023));
        float yi = (lf + rt + up + dn - deg * c) * INV_DX2;
        y[i] = yi;
        float wv = w[i];
        l0 += yi * wv;
        l1 += yi * yi;
    }
    __syncthreads();
    float s0 = block_reduce(l0, red);
    if (mode) {
        __syncthreads();
        float s1 = block_reduce(l1, red);
        if (tid == 0) { part0[blockIdx.x] = s0; part1[blockIdx.x] = s1; }
    } else {
        if (tid == 0) part0[blockIdx.x] = s0;
    }
}

// ---------------- per-link hydraulic gradient ----------------
__global__ void k_grad(const float* __restrict__ w, const float* __restrict__ q,
                       const int* __restrict__ fd, float* __restrict__ grad) {
    int l = blockIdx.x * 256 + threadIdx.x;
    if (l >= LLK) return;
    float wv = w[l];
    float pw = wv * __builtin_sqrtf(__builtin_sqrtf(wv));   // w^1.25, w > 0
    float t = q[l] / (K_FLOW * pw);
    grad[l] = t * t * (float)fd[l];
}

// ---------------- dirichlet potential at nodes + seed scalar slab ----------------
// scal: [0]=rho_prev, [1]=alpha_prev, [2]=omega_prev, [3]=rho_n current
__global__ void k_dirich(const float* __restrict__ bed, float* __restrict__ dirich,
                         float* __restrict__ scal) {
    int base = blockIdx.x * 1024 + threadIdx.x;
    #pragma unroll
    for (int k = 0; k < 4; ++k) {
        int i = base + k * 256;
        dirich[i] = RHOWG * bed[i];
    }
    if (blockIdx.x == 0 && threadIdx.x == 0) {
        scal[0] = 1.0f; scal[1] = 1.0f; scal[2] = 1.0f; scal[3] = 0.0f;
    }
}

// ---------------- r0 = b - A(0); init x,p,v; partials of dot(r0,r0) ----------------
__global__ void k_r0(const float* __restrict__ grad, const float* __restrict__ dirich,
                     const int* __restrict__ io,
                     float* __restrict__ r0, float* __restrict__ r,
                     float* __restrict__ p, float* __restrict__ v,
                     float* __restrict__ x, float* __restrict__ part0) {
    __shared__ float red[32];
    int base = blockIdx.x * 1024 + threadIdx.x;
    float acc = 0.0f;
    #pragma unroll
    for (int k = 0; k < 4; ++k) {
        int i   = base + k * 256;
        int row = i >> 10, col = i & 1023;
        float ce  = (io[i] == -1) ? dirich[i] : 0.0f;
        float sum = 0.0f, deg = 0.0f;
        if (col > 0)    { int j = i - 1;    sum += (io[j] == -1) ? dirich[j] : 0.0f; deg += 1.0f; }
        if (col < 1023) { int j = i + 1;    sum += (io[j] == -1) ? dirich[j] : 0.0f; deg += 1.0f; }
        if (row > 0)    { int j = i - 1024; sum += (io[j] == -1) ? dirich[j] : 0.0f; deg += 1.0f; }
        if (row < 1023) { int j = i + 1024; sum += (io[j] == -1) ? dirich[j] : 0.0f; deg += 1.0f; }
        float A0 = (sum - deg * ce) * INV_DX2;
        float b = 0.0f;
        if (col < 1023) b += grad[row * (CC - 1) + col];
        if (col > 0)    b -= grad[row * (CC - 1) + col - 1];
        if (row < 1023) b += grad[LH + (row << 10) + col];
        if (row > 0)    b -= grad[LH + ((row - 1) << 10) + col];
        b *= INV_DX;
        float rv = b - A0;
        r0[i] = rv; r[i] = rv; p[i] = 0.0f; v[i] = 0.0f; x[i] = 0.0f;
        acc += rv * rv;
    }
    float s = block_reduce(acc, red);
    if (threadIdx.x == 0) part0[blockIdx.x] = s;
}

// ---------------- p = r + beta*(p - omega*v); pb substituted ----------------
// rho_n summed from fresh partials; rho/alpha/omega read from scalar slab.
__global__ void k_update_p(const float* __restrict__ r, const float* __restrict__ v,
                           const int* __restrict__ io, const float* __restrict__ dirich,
                           const float* __restrict__ Prr, float* __restrict__ scal,
                           float* __restrict__ p, float* __restrict__ pb) {
    __shared__ float red[32];
    __shared__ float bc;
    float rhon = sum_partials(Prr, red, &bc);
    float rho = scal[0], alpha_p = scal[1], omega_p = scal[2];
    float beta = (rhon / (rho + EPS)) * (alpha_p / (omega_p + EPS));
    int base = blockIdx.x * 1024 + threadIdx.x;
    #pragma unroll
    for (int k = 0; k < 4; ++k) {
        int i = base + k * 256;
        float pn = r[i] + beta * (p[i] - omega_p * v[i]);
        p[i]  = pn;
        pb[i] = (io[i] == -1) ? dirich[i] : pn;
    }
    if (blockIdx.x == 0 && threadIdx.x == 0) scal[3] = rhon;   // publish rho_n
}

// ---------------- s = r - alpha*v; sb substituted ----------------
__global__ void k_update_s(const float* __restrict__ r, const float* __restrict__ v,
                           const int* __restrict__ io, const float* __restrict__ dirich,
                           const float* __restrict__ Prv, float* __restrict__ scal,
                           float* __restrict__ s, float* __restrict__ sb) {
    __shared__ float red[32];
    __shared__ float bc;
    float drv  = sum_partials(Prv, red, &bc);
    float rhon = scal[3];
    float alpha = rhon / (drv + EPS);
    int base = blockIdx.x * 1024 + threadIdx.x;
    #pragma unroll
    for (int k = 0; k < 4; ++k) {
        int i = base + k * 256;
        float sn = r[i] - alpha * v[i];
        s[i]  = sn;
        sb[i] = (io[i] == -1) ? dirich[i] : sn;
    }
    if (blockIdx.x == 0 && threadIdx.x == 0) scal[1] = alpha;  // publish alpha
}

// ---------------- x += alpha*p + omega*s; r = s - omega*t; partials dot(r0,r) ----------------
__global__ void k_update_xr(const float* __restrict__ p, const float* __restrict__ s,
                            const float* __restrict__ t, const float* __restrict__ r0,
                            const float* __restrict__ Pts, const float* __restrict__ Ptt,
                            float* __restrict__ scal,
                            float* __restrict__ x, float* __restrict__ r,
                            float* __restrict__ Pout) {
    __shared__ float red[32];
    __shared__ float bc;
    float dts = sum_partials(Pts, red, &bc);
    float dtt = sum_partials(Ptt, red, &bc);
    float alpha = scal[1];
    float omega = dts / (dtt + EPS);
    int base = blockIdx.x * 1024 + threadIdx.x;
    float acc = 0.0f;
    #pragma unroll
    for (int k = 0; k < 4; ++k) {
        int i = base + k * 256;
        x[i] = x[i] + alpha * p[i] + omega * s[i];
        float rn = s[i] - omega * t[i];
        r[i] = rn;
        acc += r0[i] * rn;
    }
    float s0 = block_reduce(acc, red);
    if (threadIdx.x == 0) Pout[blockIdx.x] = s0;
    if (blockIdx.x == 0 && threadIdx.x == 0) {
        scal[2] = omega;         // publish omega
        scal[0] = scal[3];       // rho <- rho_n (carry for next iteration's beta)
    }
}

// ---------------- effective pressure at nodes ----------------
__global__ void k_eff(const float* __restrict__ x, const float* __restrict__ dirich,
                      const float* __restrict__ ob, float* __restrict__ eff) {
    int base = blockIdx.x * 1024 + threadIdx.x;
    #pragma unroll
    for (int k = 0; k < 4; ++k) {
        int i = base + k * 256;
        float wp = fminf(fmaxf(x[i] - dirich[i], 0.0f), ob[i]);
        eff[i] = ob[i] - wp;
    }
}

// ---------------- final per-link conduit update ----------------
__global__ void k_final(const float* __restrict__ w, const float* __restrict__ q,
                        const int* __restrict__ fd, const int* __restrict__ head,
                        const int* __restrict__ tail, const float* __restrict__ eff,
                        const int* __restrict__ dtp, float* __restrict__ out /* aliases grad */) {
    int l = blockIdx.x * 256 + threadIdx.x;
    if (l >= LLK) return;
    int h = head[l], t = tail[l];
    float el = fmaxf(0.5f * (eff[h] + eff[t]), 1.0f);
    float g  = out[l];                 // gradient stored here by k_grad
    float fdf = (float)fd[l];
    float melt    = A_OPEN * q[l] * g * fdf;
    float closure = C_CLOSE * (el * el * el) * w[l];
    float dtf = (float)dtp[0];
    out[l] = fmaxf(w[l] + (melt - closure) * dtf, 0.001f);
}

extern "C" void kernel_launch(void* const* d_in, const int* in_sizes, int n_in,
                              void* d_out, int out_size, void* d_ws, size_t ws_size,
                              hipStream_t stream) {
    const float* w_in = (const float*)d_in[0];
    const float* q    = (const float*)d_in[1];
    const float* bed  = (const float*)d_in[2];
    const float* ob   = (const float*)d_in[3];
    const int*   fd   = (const int*)d_in[4];
    const int*   io   = (const int*)d_in[5];
    const int*   head = (const int*)d_in[6];
    const int*   tail = (const int*)d_in[7];
    const int*   dtp  = (const int*)d_in[8];

    float* grad = (float*)d_out;            // d_out doubles as gradient scratch
    float* f = (float*)d_ws;
    const size_t N = (size_t)NND;
    float* dirich = f;
    float* r0 = f + 1 * N;
    float* r  = f + 2 * N;
    float* p  = f + 3 * N;
    float* v  = f + 4 * N;
    float* s  = f + 5 * N;
    float* t  = f + 6 * N;
    float* x  = f + 7 * N;
    float* pb = f + 8 * N;                  // dirichlet-substituted p (reused as eff at end)
    float* sb = f + 9 * N;
    float* Prr = f + 10 * N;                // dot(r0,r) partials
    float* Prv = Prr + NPART;               // dot(r0,v) partials
    float* Pts = Prv + NPART;               // dot(t,s) partials
    float* Ptt = Pts + NPART;               // dot(t,t) partials
    float* scal = Ptt + NPART;              // [rho, alpha, omega, rho_n]

    const int gV = NND / 1024;              // 1024 blocks, 4 elems/thread
    const int gL = (LLK + 255) / 256;       // 8184
    dim3 bs(256);

    k_grad  <<<gL, bs, 0, stream>>>(w_in, q, fd, grad);
    k_dirich<<<gV, bs, 0, stream>>>(bed, dirich, scal);
    k_r0    <<<gV, bs, 0, stream>>>(grad, dirich, io, r0, r, p, v, x, Prr);

    for (int it = 0; it < 15; ++it) {
        k_update_p <<<gV, bs, 0, stream>>>(r, v, io, dirich, Prr, scal, p, pb);
        k_matvec   <<<gV, bs, 0, stream>>>(pb, r0, v, Prv, Ptt, 0);
        k_update_s <<<gV, bs, 0, stream>>>(r, v, io, dirich, Prv, scal, s, sb);
        k_matvec   <<<gV, bs, 0, stream>>>(sb, s, t, Pts, Ptt, 1);
        k_update_xr<<<gV, bs, 0, stream>>>(p, s, t, r0, Pts, Ptt, scal, x, r, Prr);
    }

    float* eff = pb; // reuse
    k_eff  <<<gV, bs, 0, stream>>>(x, dirich, ob, eff);
    k_final<<<gL, bs, 0, stream>>>(w_in, q, fd, head, tail, eff, dtp, grad);
}